// GraphSAGE_GCN_54348516164021
// MI455X (gfx1250) — compile-verified
//
#include <hip/hip_runtime.h>
#include <math.h>

#define FDIM 128
#define LN_EPS 1e-5f

typedef float v2f __attribute__((ext_vector_type(2)));
typedef float v8f __attribute__((ext_vector_type(8)));

// ---------------------------------------------------------------- zero
__global__ __launch_bounds__(256) void zero_kernel(float* __restrict__ p, long n) {
  long i = (long)blockIdx.x * blockDim.x + threadIdx.x;
  long stride = (long)gridDim.x * blockDim.x;
  for (; i < n; i += stride) p[i] = 0.0f;
}

// ------------------------------------------------- edge gather + scatter-add
// One 32-lane group (wave32) per edge: float4 gather of the 128-f32 source row
// (L2 resident), 4 f32 atomic adds per lane into s[dst], lane0 bumps degree.
__global__ __launch_bounds__(256) void scatter_kernel(
    const float* __restrict__ h, const long long* __restrict__ ei,
    float* __restrict__ s, float* __restrict__ deg, int E) {
  int eg = (int)blockIdx.x * (blockDim.x >> 5) + (threadIdx.x >> 5);
  if (eg >= E) return;
  int lane = threadIdx.x & 31;
  int src = (int)ei[eg];
  int dst = (int)ei[(long)E + eg];
  const float4* hp = (const float4*)(h + (long)src * FDIM);
  float4 v = hp[lane];                       // 32 lanes x 4 f32 = 128
  float* sp = s + (long)dst * FDIM + lane * 4;
  atomicAdd(sp + 0, v.x);
  atomicAdd(sp + 1, v.y);
  atomicAdd(sp + 2, v.z);
  atomicAdd(sp + 3, v.w);
  if (lane == 0) atomicAdd(deg + dst, 1.0f);
}

// ---------------------------------------------------------------- mean = s/deg
__global__ __launch_bounds__(256) void mean_kernel(float* __restrict__ s,
                                                   const float* __restrict__ deg,
                                                   long n) {
  long i = (long)blockIdx.x * blockDim.x + threadIdx.x;
  long stride = (long)gridDim.x * blockDim.x;
  for (; i < n; i += stride) {
    float d = deg[i >> 7];                   // i / 128
    s[i] *= 1.0f / fmaxf(d, 1.0f);
  }
}

// ------------------------------------------- fused dual GEMM via fp32 WMMA
// out[row0..row0+15, :] = A1 @ W1^T (+ A2 @ W2^T) (+ bias)
// Block = 256 threads = 8 wave32; wave w owns output cols [16w, 16w+16).
// V_WMMA_F32_16X16X4_F32, K-loop of 32 steps per operand pair.
__global__ __launch_bounds__(256) void gemm2_wmma_kernel(
    const float* __restrict__ A1, const float* __restrict__ W1,
    const float* __restrict__ A2, const float* __restrict__ W2,
    const float* __restrict__ bias, float* __restrict__ out) {
  const int row0 = blockIdx.x * 16;
  const int wave = threadIdx.x >> 5;
  const int lane = threadIdx.x & 31;
  const int n0 = wave * 16;
  const int r  = lane & 15;            // A row / B col within the tile
  const int kh = (lane >> 4) * 2;      // k sub-offset: lanes 16-31 hold k+2,k+3

  v8f acc = {};
  {
    const float* ap = A1 + (long)(row0 + r) * FDIM + kh;
    const float* bp = W1 + (long)(n0 + r) * FDIM + kh;   // B[k,n] = W[n, k]
#pragma unroll 4
    for (int k0 = 0; k0 < FDIM; k0 += 4) {
      v2f a; a.x = ap[k0]; a.y = ap[k0 + 1];
      v2f b; b.x = bp[k0]; b.y = bp[k0 + 1];
      acc = __builtin_amdgcn_wmma_f32_16x16x4_f32(
          false, a, false, b, (short)0, acc, false, false);
    }
  }
  if (A2 != nullptr) {                  // uniform (SGPR) branch: EXEC stays full
    const float* ap = A2 + (long)(row0 + r) * FDIM + kh;
    const float* bp = W2 + (long)(n0 + r) * FDIM + kh;
#pragma unroll 4
    for (int k0 = 0; k0 < FDIM; k0 += 4) {
      v2f a; a.x = ap[k0]; a.y = ap[k0 + 1];
      v2f b; b.x = bp[k0]; b.y = bp[k0 + 1];
      acc = __builtin_amdgcn_wmma_f32_16x16x4_f32(
          false, a, false, b, (short)0, acc, false, false);
    }
  }
  const int col = lane & 15;
  const float bv = (bias != nullptr) ? bias[n0 + col] : 0.0f;
  const int rb = (lane >> 4) * 8;      // D: VGPR v holds row v (+8 for hi lanes)
#pragma unroll
  for (int v = 0; v < 8; v++) {
    out[(long)(row0 + rb + v) * FDIM + n0 + col] = acc[v] + bv;
  }
}

// ----------------------------------------- graph-LayerNorm: partial reduction
__global__ __launch_bounds__(256) void ln_partial_kernel(
    const float* __restrict__ h, float* __restrict__ partials, long n) {
  __shared__ float sh1[256];
  __shared__ float sh2[256];
  float s1 = 0.0f, s2 = 0.0f;
  long i = (long)blockIdx.x * blockDim.x + threadIdx.x;
  long stride = (long)gridDim.x * blockDim.x;
  for (; i < n; i += stride) {
    float v = h[i];
    s1 += v;
    s2 += v * v;
  }
  sh1[threadIdx.x] = s1;
  sh2[threadIdx.x] = s2;
  __syncthreads();
  for (int o = 128; o > 0; o >>= 1) {
    if ((int)threadIdx.x < o) {
      sh1[threadIdx.x] += sh1[threadIdx.x + o];
      sh2[threadIdx.x] += sh2[threadIdx.x + o];
    }
    __syncthreads();
  }
  if (threadIdx.x == 0) {
    partials[2 * blockIdx.x]     = sh1[0];
    partials[2 * blockIdx.x + 1] = sh2[0];
  }
}

// ----------------------------------------- graph-LayerNorm: finalize (1 block)
__global__ __launch_bounds__(256) void ln_final_kernel(
    const float* __restrict__ partials, float* __restrict__ scal,
    int nblocks, float inv_n) {
  __shared__ float sh1[256];
  __shared__ float sh2[256];
  float s1 = 0.0f, s2 = 0.0f;
  for (int i = threadIdx.x; i < nblocks; i += 256) {
    s1 += partials[2 * i];
    s2 += partials[2 * i + 1];
  }
  sh1[threadIdx.x] = s1;
  sh2[threadIdx.x] = s2;
  __syncthreads();
  for (int o = 128; o > 0; o >>= 1) {
    if ((int)threadIdx.x < o) {
      sh1[threadIdx.x] += sh1[threadIdx.x + o];
      sh2[threadIdx.x] += sh2[threadIdx.x + o];
    }
    __syncthreads();
  }
  if (threadIdx.x == 0) {
    float m = sh1[0] * inv_n;
    float var = sh2[0] * inv_n - m * m;
    var = var < 0.0f ? 0.0f : var;
    scal[0] = m;
    scal[1] = 1.0f / (sqrtf(var) + LN_EPS);
  }
}

// ----------------------------- LN affine + PReLU (+ optional skip addend)
__global__ __launch_bounds__(256) void apply_kernel(
    const float* __restrict__ h, const float* __restrict__ scal,
    const float* __restrict__ w, const float* __restrict__ b,
    const float* __restrict__ pw, const float* __restrict__ addend,
    float* __restrict__ out, long n) {
  float m = scal[0], inv = scal[1];
  float slope = pw[0];
  long i = (long)blockIdx.x * blockDim.x + threadIdx.x;
  long stride = (long)gridDim.x * blockDim.x;
  for (; i < n; i += stride) {
    int f = (int)(i & (FDIM - 1));
    float y = (h[i] - m) * inv * w[f] + b[f];
    y = (y >= 0.0f) ? y : slope * y;
    if (addend != nullptr) y += addend[i];
    out[i] = y;
  }
}

// ================================================================ launcher
extern "C" void kernel_launch(void* const* d_in, const int* in_sizes, int n_in,
                              void* d_out, int out_size, void* d_ws, size_t ws_size,
                              hipStream_t stream) {
  const float*     x      = (const float*)d_in[0];
  const long long* ei     = (const long long*)d_in[1];   // int64 [2, E]
  const float*     Wl0    = (const float*)d_in[2];
  const float*     bl0    = (const float*)d_in[3];
  const float*     Wr0    = (const float*)d_in[4];
  const float*     ln_w0  = (const float*)d_in[5];
  const float*     ln_b0  = (const float*)d_in[6];
  const float*     pw0    = (const float*)d_in[7];
  const float*     skipW0 = (const float*)d_in[8];
  const float*     Wl1    = (const float*)d_in[9];
  const float*     bl1    = (const float*)d_in[10];
  const float*     Wr1    = (const float*)d_in[11];
  const float*     ln_w1  = (const float*)d_in[12];
  const float*     ln_b1  = (const float*)d_in[13];
  const float*     pw1    = (const float*)d_in[14];
  float*           out    = (float*)d_out;

  const int  N  = in_sizes[0] / FDIM;      // 50000
  const int  E  = in_sizes[1] / 2;         // 800000
  const long NF = (long)N * FDIM;

  // workspace layout (floats)
  float* ws    = (float*)d_ws;
  float* sbuf  = ws;                 // [N,F] segment sums -> means (in place)
  float* deg   = sbuf + NF;          // [N]
  float* conv  = deg + N;            // [N,F] conv output (pre-LN)
  float* hbuf  = conv + NF;          // [N,F] layer-1 input
  float* skipx = hbuf + NF;          // [N,F] x @ skipW0^T
  float* parts = skipx + NF;         // 2*1024
  float* scal  = parts + 2048;       // {mean, 1/(std+eps)}

  const int TB = 256;
  const dim3 blk(TB);
  const dim3 gElem(2048);                       // grid-stride elementwise
  const dim3 gScat((unsigned)((E + 7) / 8));    // 8 edges (8 waves) per block
  const dim3 gGemm((unsigned)(N / 16));         // 16 rows per block (N%16==0)
  const dim3 gLN(1024);
  const float inv_n = 1.0f / (float)NF;

  // ---------------- layer 0 (h = x) ----------------
  zero_kernel<<<gElem, blk, 0, stream>>>(sbuf, NF + N);
  scatter_kernel<<<gScat, blk, 0, stream>>>(x, ei, sbuf, deg, E);
  mean_kernel<<<gElem, blk, 0, stream>>>(sbuf, deg, NF);
  // conv = mean @ Wl0^T + x @ Wr0^T + bl0
  gemm2_wmma_kernel<<<gGemm, blk, 0, stream>>>(sbuf, Wl0, x, Wr0, bl0, conv);
  // skipx = x @ skipW0^T  (independent of conv chain)
  gemm2_wmma_kernel<<<gGemm, blk, 0, stream>>>(x, skipW0, nullptr, nullptr,
                                               nullptr, skipx);
  ln_partial_kernel<<<gLN, blk, 0, stream>>>(conv, parts, NF);
  ln_final_kernel<<<dim3(1), blk, 0, stream>>>(parts, scal, 1024, inv_n);
  // hbuf = skipx + prelu(ln(conv))
  apply_kernel<<<gElem, blk, 0, stream>>>(conv, scal, ln_w0, ln_b0, pw0,
                                          skipx, hbuf, NF);

  // ---------------- layer 1 (h = hbuf) ----------------
  zero_kernel<<<gElem, blk, 0, stream>>>(sbuf, NF + N);
  scatter_kernel<<<gScat, blk, 0, stream>>>(hbuf, ei, sbuf, deg, E);
  mean_kernel<<<gElem, blk, 0, stream>>>(sbuf, deg, NF);
  gemm2_wmma_kernel<<<gGemm, blk, 0, stream>>>(sbuf, Wl1, hbuf, Wr1, bl1, conv);
  ln_partial_kernel<<<gLN, blk, 0, stream>>>(conv, parts, NF);
  ln_final_kernel<<<dim3(1), blk, 0, stream>>>(parts, scal, 1024, inv_n);
  apply_kernel<<<gElem, blk, 0, stream>>>(conv, scal, ln_w1, ln_b1, pw1,
                                          nullptr, out, NF);
}